// RandLANet_43696997269973
// MI455X (gfx1250) — compile-verified
//
#include <hip/hip_runtime.h>
#include <math.h>

// ---------------------------------------------------------------------------
// MI455X (gfx1250) RandLA-Net forward.
//  - KNN pairwise-distance tiles + all 1x1-conv GEMMs via V_WMMA_F32_16X16X4_F32
//  - wave32, one 16x32 output tile per wave in the conv (2 WMMAs share one
//    A operand per K-step), EXEC all-1s at every WMMA
//  - LDS staging for per-row top-K selection in the KNN kernel
// ---------------------------------------------------------------------------

typedef __attribute__((ext_vector_type(2))) float v2f;
typedef __attribute__((ext_vector_type(8))) float v8f;

#define BATCH   2
#define NPTS    8192
#define BN_INV  0.9999995f   /* 1/sqrt(1+1e-6) */

__device__ __forceinline__ float actf(float v, int act) {
  if (act == 1) return v > 0.f ? v : 0.f;          // relu
  if (act == 2) return v > 0.f ? v : 0.2f  * v;    // leaky 0.2
  if (act == 3) return v > 0.f ? v : 0.01f * v;    // leaky 0.01
  return v;
}

// ---------------------------------------------------------------------------
// Generic pointwise conv (einsum 'oc,bcm->bom') with fused bias/BN/add/act.
// One wave computes a 16(Cout) x 32(M) tile; K-loop over Cin in steps of 4
// using two v_wmma_f32_16x16x4_f32 that share the A (weight) operand.
// A (16x4): lane l<16 row M=l; V0 = K(0|2 by lane-half), V1 = K(1|3).
// B (4x16): lane l<16 col N=l; V0 = K(0|2 by lane-half), V1 = K(1|3).
// D (16x16): VGPR v, lanes 0-15 -> (M=v, N=l); lanes 16-31 -> (M=v+8, N=l).
// ---------------------------------------------------------------------------
__global__ void conv_wmma_kernel(const float* __restrict__ x, float* __restrict__ y,
                                 const float* __restrict__ W, const float* __restrict__ bias,
                                 const float* __restrict__ g, const float* __restrict__ bt,
                                 const float* __restrict__ add,
                                 int Cin, int Cout, int M, int inRS,
                                 long inBS, long outBS, long addBS,
                                 int act, int tO, int tM, int nB)
{
  int wave = (int)(((long)blockIdx.x * blockDim.x + threadIdx.x) >> 5);
  int lane = threadIdx.x & 31;
  int tpb  = tO * tM;
  int b    = wave / tpb;
  if (b >= nB) return;                       // wave-uniform exit
  int t  = wave - b * tpb;
  int o0 = (t / tM) * 16;
  int m0 = (t % tM) * 32;

  const float* xb = x + (size_t)b * inBS;
  int half = lane >> 4;
  int l    = lane & 15;
  int arow  = o0 + l;
  int bcol0 = m0 + l;
  int bcol1 = m0 + 16 + l;

  const bool aok  = arow  < Cout;
  const bool b0ok = bcol0 < M;
  const bool b1ok = bcol1 < M;
  const float* wr = W + (size_t)arow * Cin;

  v8f acc0 = {0.f, 0.f, 0.f, 0.f, 0.f, 0.f, 0.f, 0.f};
  v8f acc1 = {0.f, 0.f, 0.f, 0.f, 0.f, 0.f, 0.f, 0.f};
  for (int c = 0; c < Cin; c += 4) {
    int k0 = c + 2 * half;
    int k1 = k0 + 1;
    bool k0ok = k0 < Cin, k1ok = k1 < Cin;
    v2f av, bv0, bv1;
    av.x  = (aok  && k0ok) ? wr[k0] : 0.f;
    av.y  = (aok  && k1ok) ? wr[k1] : 0.f;
    bv0.x = (b0ok && k0ok) ? xb[(size_t)k0 * inRS + bcol0] : 0.f;
    bv0.y = (b0ok && k1ok) ? xb[(size_t)k1 * inRS + bcol0] : 0.f;
    bv1.x = (b1ok && k0ok) ? xb[(size_t)k0 * inRS + bcol1] : 0.f;
    bv1.y = (b1ok && k1ok) ? xb[(size_t)k1 * inRS + bcol1] : 0.f;
    acc0 = __builtin_amdgcn_wmma_f32_16x16x4_f32(false, av, false, bv0,
                                                 (short)0, acc0, false, false);
    acc1 = __builtin_amdgcn_wmma_f32_16x16x4_f32(false, av, false, bv1,
                                                 (short)0, acc1, false, false);
  }

  float* yb = y + (size_t)b * outBS;
  const float* ab = add ? (add + (size_t)b * addBS) : nullptr;
#pragma unroll
  for (int v = 0; v < 8; ++v) {
    int row = o0 + v + 8 * half;
    if (row < Cout) {
      float bb = bias[row];
      float scale = g ? (g[row] * BN_INV) : 1.f;
      float shift = g ? bt[row] : 0.f;
      if (b0ok) {
        float val = (acc0[v] + bb) * scale + shift;
        if (ab) val += ab[(size_t)row * M + bcol0];
        yb[(size_t)row * M + bcol0] = actf(val, act);
      }
      if (b1ok) {
        float val = (acc1[v] + bb) * scale + shift;
        if (ab) val += ab[(size_t)row * M + bcol1];
        yb[(size_t)row * M + bcol1] = actf(val, act);
      }
    }
  }
}

// ---------------------------------------------------------------------------
// KNN: one wave (block) handles 16 target points. Per 16-source tile:
// WMMA computes the 16x16 dot tile, LDS stages it, lanes 0-15 each maintain
// a register-resident running top-16 (replace-max) over their target row.
// Outputs the K smallest (sorted) with dist = sqrt(max(d2,0)).
// ---------------------------------------------------------------------------
__global__ void knn_kernel(const float* __restrict__ coords4, int n_src, int n_tgt,
                           int K, int* __restrict__ idx_out, float* __restrict__ dist_out)
{
  __shared__ float tile[16][17];
  __shared__ float snorm_s[16];

  int tilesPerBatch = n_tgt >> 4;
  int b    = blockIdx.x / tilesPerBatch;
  int t0   = (blockIdx.x % tilesPerBatch) << 4;
  int lane = threadIdx.x;
  int half = lane >> 4;
  int l    = lane & 15;

  const float* cb = coords4 + (size_t)b * NPTS * 4;

  v2f av;
  av.x = cb[(t0 + l) * 4 + 2 * half];
  av.y = cb[(t0 + l) * 4 + 2 * half + 1];
  float tx = cb[(t0 + l) * 4 + 0];
  float ty = cb[(t0 + l) * 4 + 1];
  float tz = cb[(t0 + l) * 4 + 2];
  float tnorm = tx * tx + ty * ty + tz * tz;

  float bd[16];
  int   bi[16];
#pragma unroll
  for (int j = 0; j < 16; ++j) { bd[j] = 3.4e38f; bi[j] = 0; }
  float curMax = 3.4e38f;

  for (int s0 = 0; s0 < n_src; s0 += 16) {
    if (s0 + 16 < n_src)
      __builtin_prefetch(&cb[(s0 + 16 + l) * 4], 0, 1);

    v2f bv;
    bv.x = cb[(s0 + l) * 4 + 2 * half];
    bv.y = cb[(s0 + l) * 4 + 2 * half + 1];
    v8f d = {0.f, 0.f, 0.f, 0.f, 0.f, 0.f, 0.f, 0.f};
    d = __builtin_amdgcn_wmma_f32_16x16x4_f32(false, av, false, bv,
                                              (short)0, d, false, false);
    if (lane < 16) {
      float sx = cb[(s0 + l) * 4 + 0];
      float sy = cb[(s0 + l) * 4 + 1];
      float sz = cb[(s0 + l) * 4 + 2];
      snorm_s[l] = sx * sx + sy * sy + sz * sz;
    }
#pragma unroll
    for (int v = 0; v < 8; ++v) tile[v + 8 * half][l] = d[v];
    __syncthreads();

    if (lane < 16) {
      for (int j = 0; j < 16; ++j) {
        float d2 = fmaxf(tnorm - 2.f * tile[l][j] + snorm_s[j], 0.f);
        if (d2 < curMax) {
          bool rep = false;
#pragma unroll
          for (int q = 0; q < 16; ++q)
            if (!rep && bd[q] == curMax) { bd[q] = d2; bi[q] = s0 + j; rep = true; }
          curMax = bd[0];
#pragma unroll
          for (int q = 1; q < 16; ++q) curMax = fmaxf(curMax, bd[q]);
        }
      }
    }
    __syncthreads();
  }

  if (lane < 16) {
    long base = ((long)b * n_tgt + (t0 + l)) * K;
    for (int i = 0; i < K; ++i) {
      float mn = 3.4e38f;
#pragma unroll
      for (int q = 0; q < 16; ++q) mn = fminf(mn, bd[q]);
      int sel = -1;
#pragma unroll
      for (int q = 0; q < 16; ++q)
        if (sel < 0 && bd[q] == mn) { sel = bi[q]; bd[q] = 3.4e38f; }
      idx_out[base + i]  = sel;
      dist_out[base + i] = sqrtf(mn);
    }
  }
}

// --------------------------- small helper kernels ---------------------------

__global__ void mean_kernel(const float* __restrict__ inp, float* __restrict__ mean)
{
  __shared__ float sm[3][256];
  int b = blockIdx.x, t = threadIdx.x;
  float s0 = 0.f, s1 = 0.f, s2 = 0.f;
  for (int i = t; i < NPTS; i += 256) {
    const float* p = inp + ((size_t)b * NPTS + i) * 4;
    s0 += p[0]; s1 += p[1]; s2 += p[2];
  }
  sm[0][t] = s0; sm[1][t] = s1; sm[2][t] = s2;
  __syncthreads();
  for (int off = 128; off > 0; off >>= 1) {
    if (t < off) {
      sm[0][t] += sm[0][t + off];
      sm[1][t] += sm[1][t + off];
      sm[2][t] += sm[2][t + off];
    }
    __syncthreads();
  }
  if (t == 0) {
    mean[b * 4 + 0] = sm[0][0] / (float)NPTS;
    mean[b * 4 + 1] = sm[1][0] / (float)NPTS;
    mean[b * 4 + 2] = sm[2][0] / (float)NPTS;
  }
}

// normalize coords, fc 4->8, bn0, tanh, apply permutation (gather)
__global__ void pre_kernel(const float* __restrict__ inp, const float* __restrict__ mean,
                           const int* __restrict__ perm,
                           const float* __restrict__ fcW, const float* __restrict__ fcb,
                           const float* __restrict__ bn0g, const float* __restrict__ bn0b,
                           float* __restrict__ coords4, float* __restrict__ x0)
{
  long gid = (long)blockIdx.x * blockDim.x + threadIdx.x;
  if (gid >= (long)BATCH * NPTS) return;
  int b = (int)(gid / NPTS), j = (int)(gid % NPTS);
  int p = perm[j];
  const float* ip = inp + ((size_t)b * NPTS + p) * 4;
  float c0 = (ip[0] - mean[b * 4 + 0]) * 0.1f;
  float c1 = (ip[1] - mean[b * 4 + 1]) * 0.1f;
  float c2 = (ip[2] - mean[b * 4 + 2]) * 0.1f;
  float w  = ip[3];
  float* cp = coords4 + ((size_t)b * NPTS + j) * 4;
  cp[0] = c0; cp[1] = c1; cp[2] = c2; cp[3] = 0.f;
#pragma unroll
  for (int o = 0; o < 8; ++o) {
    float v = fcb[o] + c0 * fcW[o * 4 + 0] + c1 * fcW[o * 4 + 1]
                     + c2 * fcW[o * 4 + 2] + w  * fcW[o * 4 + 3];
    v = v * (bn0g[o] * BN_INV) + bn0b[o];
    x0[(size_t)b * 8 * NPTS + (size_t)o * NPTS + j] = tanhf(v);
  }
}

// build the 10-channel LSE input: [center(3), nb(3), center-nb(3), dist(1)]
__global__ void lse_gather_kernel(const float* __restrict__ coords4,
                                  const int* __restrict__ idx, const float* __restrict__ dist,
                                  float* __restrict__ cat, int n, int k)
{
  long gid = (long)blockIdx.x * blockDim.x + threadIdx.x;
  long tot = (long)BATCH * n * k;
  if (gid >= tot) return;
  int  b = (int)(gid / ((long)n * k));
  long r = gid % ((long)n * k);
  int  i = (int)(r / k);
  int  j = (int)(r % k);
  const float* cb = coords4 + (size_t)b * NPTS * 4;
  int q = idx[((size_t)b * n + i) * k + j];
  float cx = cb[i * 4 + 0], cy = cb[i * 4 + 1], cz = cb[i * 4 + 2];
  float nx = cb[q * 4 + 0], ny = cb[q * 4 + 1], nz = cb[q * 4 + 2];
  size_t M = (size_t)n * k;
  float* cp = cat + (size_t)b * 10 * M + (size_t)r;
  cp[0 * M] = cx;      cp[1 * M] = cy;      cp[2 * M] = cz;
  cp[3 * M] = nx;      cp[4 * M] = ny;      cp[5 * M] = nz;
  cp[6 * M] = cx - nx; cp[7 * M] = cy - ny; cp[8 * M] = cz - nz;
  cp[9 * M] = dist[((size_t)b * n + i) * k + j];
}

// broadcast feats (B,h,n) into channels [h,2h) of enc buffer (B,2h,n*k)
__global__ void bcast_kernel(const float* __restrict__ feats, float* __restrict__ enc,
                             int h, int n, int k)
{
  long gid = (long)blockIdx.x * blockDim.x + threadIdx.x;
  long tot = (long)BATCH * h * n * k;
  if (gid >= tot) return;
  int  b  = (int)(gid / ((long)h * n * k));
  long r  = gid % ((long)h * n * k);
  int  c  = (int)(r / ((long)n * k));
  long rr = r % ((long)n * k);
  int  i  = (int)(rr / k);
  enc[(size_t)b * (2 * h) * n * k + (size_t)(h + c) * n * k + (size_t)rr] =
      feats[(size_t)b * h * n + (size_t)c * n + i];
}

// softmax over k of score, weighted sum of enc -> (B,C,n)
__global__ void softmax_pool_kernel(const float* __restrict__ enc, const float* __restrict__ sc,
                                    float* __restrict__ out, int C, int n, int k)
{
  long gid = (long)blockIdx.x * blockDim.x + threadIdx.x;
  long tot = (long)BATCH * C * n;
  if (gid >= tot) return;
  int b = (int)(gid / ((long)C * n));
  long r = gid % ((long)C * n);
  int c = (int)(r / n), i = (int)(r % n);
  size_t M = (size_t)n * k;
  const float* sp = sc  + ((size_t)b * C + c) * M + (size_t)i * k;
  const float* ep = enc + ((size_t)b * C + c) * M + (size_t)i * k;
  float mx = sp[0];
  for (int j = 1; j < k; ++j) mx = fmaxf(mx, sp[j]);
  float se = 0.f, acc = 0.f;
  for (int j = 0; j < k; ++j) {
    float e = expf(sp[j] - mx);
    se += e; acc += e * ep[j];
  }
  out[((size_t)b * C + c) * n + i] = acc / se;
}

__global__ void copy_rows_kernel(const float* __restrict__ in, float* __restrict__ out,
                                 int C, int M, int inRS, long inBS, int outRS, long outBS)
{
  long gid = (long)blockIdx.x * blockDim.x + threadIdx.x;
  long tot = (long)BATCH * C * M;
  if (gid >= tot) return;
  int b = (int)(gid / ((long)C * M));
  long r = gid % ((long)C * M);
  int c = (int)(r / M), m = (int)(r % M);
  out[(size_t)b * outBS + (size_t)c * outRS + m] =
      in[(size_t)b * inBS + (size_t)c * inRS + m];
}

// inverse-distance weighted 3-NN upsampling into channels [0,C) of concat buf
__global__ void interp_kernel(const float* __restrict__ x, const int* __restrict__ idx,
                              const float* __restrict__ dist, float* __restrict__ out,
                              int C, int n_down, int n_up, long outBS)
{
  long gid = (long)blockIdx.x * blockDim.x + threadIdx.x;
  long tot = (long)BATCH * C * n_up;
  if (gid >= tot) return;
  int b = (int)(gid / ((long)C * n_up));
  long r = gid % ((long)C * n_up);
  int c = (int)(r / n_up), i = (int)(r % n_up);
  const int*   ip = idx  + ((size_t)b * n_up + i) * 3;
  const float* dp = dist + ((size_t)b * n_up + i) * 3;
  float w0 = 1.f / (dp[0] + 1e-8f);
  float w1 = 1.f / (dp[1] + 1e-8f);
  float w2 = 1.f / (dp[2] + 1e-8f);
  float s = w0 + w1 + w2;
  const float* xb = x + ((size_t)b * C + c) * n_down;
  out[(size_t)b * outBS + (size_t)c * n_up + i] =
      (w0 * xb[ip[0]] + w1 * xb[ip[1]] + w2 * xb[ip[2]]) / s;
}

// scatter through permutation: out[:, :, perm[i]] = in[:, :, i]
__global__ void unperm_kernel(const float* __restrict__ in, const int* __restrict__ perm,
                              float* __restrict__ out, int C)
{
  long gid = (long)blockIdx.x * blockDim.x + threadIdx.x;
  long tot = (long)BATCH * C * NPTS;
  if (gid >= tot) return;
  int b = (int)(gid / ((long)C * NPTS));
  long r = gid % ((long)C * NPTS);
  int c = (int)(r / NPTS), i = (int)(r % NPTS);
  out[((size_t)b * C + c) * NPTS + perm[i]] = in[((size_t)b * C + c) * NPTS + i];
}

// ------------------------------- host side ----------------------------------

struct ConvP { const float *W, *b, *bt, *g; };

static inline long cdivl(long a, long b) { return (a + b - 1) / b; }

static void run_conv(hipStream_t stream, const float* x, float* y,
                     const ConvP& p, const float* add,
                     int Cin, int Cout, int M, int inRS,
                     long inBS, long outBS, long addBS, int act)
{
  int tO = (Cout + 15) / 16;
  int tM = (M + 31) / 32;                // 16x32 output tile per wave
  long waves  = (long)BATCH * tO * tM;
  int  wpb    = 4;                       // 128 threads = 4 waves per block
  long blocks = cdivl(waves, wpb);
  conv_wmma_kernel<<<dim3((unsigned)blocks), dim3(wpb * 32), 0, stream>>>(
      x, y, p.W, p.b, p.g, p.bt, add,
      Cin, Cout, M, inRS, inBS, outBS, addBS, act, tO, tM, BATCH);
}

#define EW_LAUNCH(kern, tot, ...)                                                \
  kern<<<dim3((unsigned)(((long)(tot) + 255) / 256)), dim3(256), 0, stream>>>(   \
      __VA_ARGS__)

extern "C" void kernel_launch(void* const* d_in, const int* in_sizes, int n_in,
                              void* d_out, int out_size, void* d_ws, size_t ws_size,
                              hipStream_t stream)
{
  (void)in_sizes; (void)n_in; (void)out_size; (void)ws_size;

  // ------------------ unpack params (JAX pytree sorted-key order) -----------
  int ii = 0;
  auto F = [&]() -> const float* { return (const float*)d_in[ii++]; };

  const float* inp  = F();
  const float* bn0b = F();
  const float* bn0g = F();

  ConvP dec[4];
  for (int d = 0; d < 4; ++d) { dec[d].W = F(); dec[d].b = F(); dec[d].bt = F(); dec[d].g = F(); }

  ConvP lse1[4], lse2[4], mlp1e[4], mlp2e[4], p1o[4], p1s[4], p2o[4], p2s[4], shc[4];
  for (int l = 0; l < 4; ++l) {
    lse1[l].W  = F(); lse1[l].b  = F(); lse1[l].bt = F(); lse1[l].g = F();
    lse2[l].W  = F(); lse2[l].b  = F(); lse2[l].bt = F(); lse2[l].g = F();
    mlp1e[l].W = F(); mlp1e[l].b = F(); mlp1e[l].bt = nullptr; mlp1e[l].g = nullptr;
    mlp2e[l].W = F(); mlp2e[l].b = F(); mlp2e[l].bt = nullptr; mlp2e[l].g = nullptr;
    p1o[l].W = F(); p1o[l].b = F(); p1o[l].bt = F(); p1o[l].g = F();
    p1s[l].W = F(); p1s[l].b = F(); p1s[l].bt = nullptr; p1s[l].g = nullptr;
    p2o[l].W = F(); p2o[l].b = F(); p2o[l].bt = F(); p2o[l].g = F();
    p2s[l].W = F(); p2s[l].b = F(); p2s[l].bt = nullptr; p2s[l].g = nullptr;
    shc[l].W = F(); shc[l].b = F(); shc[l].bt = F(); shc[l].g = F();
  }
  const float* fcW = F();
  const float* fcb = F();
  ConvP fce[3];
  fce[0].W = F(); fce[0].b = F(); fce[0].bt = F(); fce[0].g = F();
  fce[1].W = F(); fce[1].b = F(); fce[1].bt = F(); fce[1].g = F();
  fce[2].W = F(); fce[2].b = F(); fce[2].bt = nullptr; fce[2].g = nullptr;
  ConvP midp; midp.W = F(); midp.b = F(); midp.bt = nullptr; midp.g = nullptr;
  const int* perm = (const int*)d_in[ii++];

  // ------------------ workspace bump allocator ------------------------------
  char*  base = (char*)d_ws;
  size_t off  = 0;
  auto alloc  = [&](size_t bytes) -> void* {
    void* p = base + off;
    off = (off + bytes + 255) & ~(size_t)255;
    return p;
  };
  auto allocF = [&](size_t n) -> float* { return (float*)alloc(n * 4); };
  auto allocI = [&](size_t n) -> int*   { return (int*)alloc(n * 4); };

  const int nl[4]  = {8192, 2048, 512, 128};   // points per encoder level
  const int dinl[4]= {8, 32, 128, 256};
  const int hl[4]  = {8, 32, 64, 128};
  const int dl[4]  = {16, 64, 128, 256};       // dout = 2h

  float* meanB   = allocF((size_t)BATCH * 4);
  float* coords4 = allocF((size_t)BATCH * NPTS * 4);
  float* x0      = allocF((size_t)BATCH * 8 * NPTS);
  int*   idxE[4]; float* distE[4];
  for (int l = 0; l < 4; ++l) {
    idxE[l]  = allocI((size_t)BATCH * nl[l] * 16);
    distE[l] = allocF((size_t)BATCH * nl[l] * 16);
  }
  const int ndn[4] = {32, 128, 512, 2048};     // decoder n_down
  int*   idxD[4]; float* distD[4];
  for (int d = 0; d < 4; ++d) {
    idxD[d]  = allocI((size_t)BATCH * (4 * ndn[d]) * 3);
    distD[d] = allocF((size_t)BATCH * (4 * ndn[d]) * 3);
  }
  float* stackB[4];
  for (int l = 0; l < 4; ++l) stackB[l] = allocF((size_t)BATCH * 2 * dl[l] * nl[l]);

  float* cat     = allocF((size_t)BATCH * 10 * NPTS * 16);
  float* enc     = allocF((size_t)BATCH * 16 * NPTS * 16);
  float* sc      = allocF((size_t)BATCH * 16 * NPTS * 16);
  float* psum    = allocF((size_t)BATCH * 131072);
  float* tmlp1   = allocF((size_t)BATCH * 65536);
  float* tpool1  = allocF((size_t)BATCH * 65536);
  float* tpool2  = allocF((size_t)BATCH * 131072);
  float* shcT    = allocF((size_t)BATCH * 262144);
  float* xmid    = allocF((size_t)BATCH * 512 * 32);
  float* dconcat = allocF((size_t)BATCH * 524288);
  float* dxA     = allocF((size_t)BATCH * 65536);
  float* dxB     = allocF((size_t)BATCH * 65536);
  float* unp     = allocF((size_t)BATCH * 8 * NPTS);
  float* fct1    = allocF((size_t)BATCH * 64 * NPTS);
  float* fct2    = allocF((size_t)BATCH * 32 * NPTS);

  // ------------------ preprocess -------------------------------------------
  mean_kernel<<<dim3(BATCH), dim3(256), 0, stream>>>(inp, meanB);
  EW_LAUNCH(pre_kernel, (long)BATCH * NPTS,
            inp, meanB, perm, fcW, fcb, bn0g, bn0b, coords4, x0);

  // ------------------ encoder ----------------------------------------------
  const float* xin   = x0;
  int          xinRS = NPTS;
  long         xinBS = (long)8 * NPTS;

  for (int l = 0; l < 4; ++l) {
    int n = nl[l], h = hl[l], dout = dl[l], din = dinl[l];
    long M2 = (long)n * 16;

    knn_kernel<<<dim3((unsigned)(BATCH * (n / 16))), dim3(32), 0, stream>>>(
        coords4, n, n, 16, idxE[l], distE[l]);

    // mlp1: din->h, leaky 0.2 (strided read of level input)
    run_conv(stream, xin, tmlp1, mlp1e[l], nullptr,
             din, h, n, xinRS, xinBS, (long)h * n, 0, 2);

    // LSE input (shared between lse1 and lse2)
    EW_LAUNCH(lse_gather_kernel, (long)BATCH * M2, coords4, idxE[l], distE[l], cat, n, 16);

    // lse1 conv 10->h (bn+relu) into enc[0:h), broadcast mlp1 feats into [h:2h)
    run_conv(stream, cat, enc, lse1[l], nullptr,
             10, h, (int)M2, (int)M2, 10L * M2, (long)dout * M2, 0, 1);
    EW_LAUNCH(bcast_kernel, (long)BATCH * h * M2, tmlp1, enc, h, n, 16);

    // pool1: score conv, softmax-pool, out conv (dout->h, bn+relu)
    run_conv(stream, enc, sc, p1s[l], nullptr,
             dout, dout, (int)M2, (int)M2, (long)dout * M2, (long)dout * M2, 0, 0);
    EW_LAUNCH(softmax_pool_kernel, (long)BATCH * dout * n, enc, sc, psum, dout, n, 16);
    run_conv(stream, psum, tpool1, p1o[l], nullptr,
             dout, h, n, n, (long)dout * n, (long)h * n, 0, 1);

    // lse2 (same cat, feats = pool1 out)
    run_conv(stream, cat, enc, lse2[l], nullptr,
             10, h, (int)M2, (int)M2, 10L * M2, (long)dout * M2, 0, 1);
    EW_LAUNCH(bcast_kernel, (long)BATCH * h * M2, tpool1, enc, h, n, 16);

    // pool2 (out conv dout->dout, bn+relu)
    run_conv(stream, enc, sc, p2s[l], nullptr,
             dout, dout, (int)M2, (int)M2, (long)dout * M2, (long)dout * M2, 0, 0);
    EW_LAUNCH(softmax_pool_kernel, (long)BATCH * dout * n, enc, sc, psum, dout, n, 16);
    run_conv(stream, psum, tpool2, p2o[l], nullptr,
             dout, dout, n, n, (long)dout * n, (long)dout * n, 0, 1);

    // shortcut (din->2*dout, bn, no act) from level input
    run_conv(stream, xin, shcT, shc[l], nullptr,
             din, 2 * dout, n, xinRS, xinBS, (long)2 * dout * n, 0, 0);
    // mlp2 (dout->2*dout) + shortcut, leaky 0.01 -> stack buffer
    run_conv(stream, tpool2, stackB[l], mlp2e[l], shcT,
             dout, 2 * dout, n, n, (long)dout * n,
             (long)2 * dout * n, (long)2 * dout * n, 3);

    xin = stackB[l]; xinRS = n; xinBS = (long)2 * dout * n;
  }

  // ------------------ mid (512->512, relu) on first 32 columns -------------
  run_conv(stream, stackB[3], xmid, midp, nullptr,
           512, 512, 32, 128, 512L * 128, 512L * 32, 0, 1);

  // ------------------ decoder ----------------------------------------------
  const float* xc     = xmid;
  const int    Cd[4]  = {512, 256, 128, 32};
  const int    CoD[4] = {256, 128, 32, 8};
  float*       dxbuf[2] = {dxA, dxB};

  for (int d = 0; d < 4; ++d) {
    int n_down = ndn[d], n_up = 4 * n_down, C = Cd[d];

    knn_kernel<<<dim3((unsigned)(BATCH * (n_up / 16))), dim3(32), 0, stream>>>(
        coords4, n_down, n_up, 3, idxD[d], distD[d]);

    long cBS = (long)2 * C * n_up;
    EW_LAUNCH(interp_kernel, (long)BATCH * C * n_up,
              xc, idxD[d], distD[d], dconcat, C, n_down, n_up, cBS);
    // skip connection into channels [C, 2C)
    EW_LAUNCH(copy_rows_kernel, (long)BATCH * C * n_up,
              stackB[3 - d], dconcat + (size_t)C * n_up,
              C, n_up, n_up, (long)C * n_up, n_up, cBS);

    float* xn = dxbuf[d & 1];
    run_conv(stream, dconcat, xn, dec[d], nullptr,
             2 * C, CoD[d], n_up, n_up, cBS, (long)CoD[d] * n_up, 0, 1);
    xc = xn;
  }

  // ------------------ un-permute + fc_end ----------------------------------
  EW_LAUNCH(unperm_kernel, (long)BATCH * 8 * NPTS, xc, perm, unp, 8);

  run_conv(stream, unp,  fct1, fce[0], nullptr,
           8, 64, NPTS, NPTS, 8L * NPTS, 64L * NPTS, 0, 1);
  run_conv(stream, fct1, fct2, fce[1], nullptr,
           64, 32, NPTS, NPTS, 64L * NPTS, 32L * NPTS, 0, 1);
  run_conv(stream, fct2, (float*)d_out, fce[2], nullptr,
           32, 10, NPTS, NPTS, 32L * NPTS, 10L * NPTS, 0, 0);
}